// TensorProduct_6244882448674
// MI455X (gfx1250) — compile-verified
//
#include <hip/hip_runtime.h>

// out[b,n,f] = sum_{l,m} x1[b,l,f] * x2[b,m,f] * cg[l,m,n]
// Cast as GEMM: W[n, k=9l+m] (25x81, pad 32x84)  x  Y[k, col=(b,f)] -> out[n, col]
// using V_WMMA_F32_16X16X4_F32 (full fp32, matches reference numerics).

typedef __attribute__((ext_vector_type(2))) float v2f;
typedef __attribute__((ext_vector_type(8))) float v8f;

#define D1      9      // (LMAX1+1)^2
#define D3      25     // (LMAX_OUT+1)^2
#define MULCH   512    // channel dim 'f'
#define KDIM    81     // 9*9 (l,m) pairs
#define KSTEPS  21     // ceil(84/4) K-steps of 4
#define K1_FIRST 10    // n=16..24 tile fed only by l1=l2=2 block => k>=40 => steps 10..20
#define K1_STEPS 11

__global__ __launch_bounds__(256) void tp_wmma_kernel(
    const float* __restrict__ x1,
    const float* __restrict__ x2,
    const float* __restrict__ cg,
    float* __restrict__ out,
    int nTiles)
{
    const int lane = threadIdx.x & 31;
    const int half = lane >> 4;     // selects K pair {0,1} vs {2,3} in A/B fragments
    const int colL = lane & 15;     // column within 16-wide tile / row M for A

    const int wavesPerBlock = blockDim.x >> 5;
    const int waveId = blockIdx.x * wavesPerBlock + (threadIdx.x >> 5);
    const int nWaves = gridDim.x * wavesPerBlock;

    // ---- Load A = W (CG matrix) fragments once per wave; amortized over tiles ----
    // A 16x4 f32 layout: lane L holds row M = L%16; VGPR0 = K = half*2, VGPR1 = K = half*2+1.
    v2f a0[KSTEPS];     // rows n = 0..15
#pragma unroll
    for (int kk = 0; kk < KSTEPS; ++kk) {
        const int n = colL;
        const int k = 4 * kk + half * 2;
        a0[kk].x = (k     < KDIM) ? cg[k * D3 + n]       : 0.0f;
        a0[kk].y = (k + 1 < KDIM) ? cg[(k + 1) * D3 + n] : 0.0f;
    }
    v2f a1[K1_STEPS];   // rows n = 16..24 (pad to 31)
#pragma unroll
    for (int i = 0; i < K1_STEPS; ++i) {
        const int kk = K1_FIRST + i;
        const int n = 16 + colL;
        const int k = 4 * kk + half * 2;
        a1[i].x = (n < D3 && k     < KDIM) ? cg[k * D3 + n]       : 0.0f;
        a1[i].y = (n < D3 && k + 1 < KDIM) ? cg[(k + 1) * D3 + n] : 0.0f;
    }

    const int tilesPerBatch = MULCH / 16;   // 32

    for (int tile = waveId; tile < nTiles; tile += nWaves) {
        const int b  = tile / tilesPerBatch;
        const int fb = (tile - b * tilesPerBatch) * 16;

        // Each lane services column colL; both half-waves load the same 16
        // channels (coalesces to one 64B line per row, no extra HBM traffic).
        const float* p1 = x1 + (size_t)b * (D1 * MULCH) + fb + colL;
        const float* p2 = x2 + (size_t)b * (D1 * MULCH) + fb + colL;
        float xv[D1], yv[D1];
#pragma unroll
        for (int l = 0; l < D1; ++l) {
            xv[l] = p1[l * MULCH];
            yv[l] = p2[l * MULCH];
        }

        v8f acc0 = {0.f, 0.f, 0.f, 0.f, 0.f, 0.f, 0.f, 0.f};
        v8f acc1 = {0.f, 0.f, 0.f, 0.f, 0.f, 0.f, 0.f, 0.f};

#pragma unroll
        for (int kk = 0; kk < KSTEPS; ++kk) {
            const int k0 = 4 * kk, k1 = k0 + 1, k2 = k0 + 2, k3 = k0 + 3;
            // Y[k, col] = x1[k/9] * x2[k%9]; indices constant-folded by unroll.
            const float pr0 =                xv[(k0 / 9) % D1] * yv[k0 % 9];
            const float pr1 = (k1 < KDIM) ? (xv[(k1 / 9) % D1] * yv[k1 % 9]) : 0.0f;
            const float pr2 = (k2 < KDIM) ? (xv[(k2 / 9) % D1] * yv[k2 % 9]) : 0.0f;
            const float pr3 = (k3 < KDIM) ? (xv[(k3 / 9) % D1] * yv[k3 % 9]) : 0.0f;
            // B 4x16 f32 fragment: VGPR0 -> K = half*2, VGPR1 -> K = half*2+1.
            v2f bvec;
            bvec.x = half ? pr2 : pr0;
            bvec.y = half ? pr3 : pr1;

            acc0 = __builtin_amdgcn_wmma_f32_16x16x4_f32(
                false, a0[kk], false, bvec, (short)0, acc0, false, false);
            if (kk >= K1_FIRST) {
                acc1 = __builtin_amdgcn_wmma_f32_16x16x4_f32(
                    false, a1[kk - K1_FIRST], false, bvec, (short)0, acc1, false, false);
            }
        }

        // D 16x16 f32 layout: VGPR r -> row r (lanes 0-15) / row r+8 (lanes 16-31).
        float* po = out + (size_t)b * (D3 * MULCH) + fb + colL;
#pragma unroll
        for (int r = 0; r < 8; ++r) {
            const int n0 = r + half * 8;            // rows 0..15, always valid
            po[(size_t)n0 * MULCH] = acc0[r];
            const int n1 = 16 + r + half * 8;       // rows 16..31, valid if < 25
            if (n1 < D3) po[(size_t)n1 * MULCH] = acc1[r];
        }
    }
}

extern "C" void kernel_launch(void* const* d_in, const int* in_sizes, int n_in,
                              void* d_out, int out_size, void* d_ws, size_t ws_size,
                              hipStream_t stream) {
    const float* x1 = (const float*)d_in[0];
    const float* x2 = (const float*)d_in[1];
    const float* cg = (const float*)d_in[2];
    float* out = (float*)d_out;

    const int batch = in_sizes[0] / (D1 * MULCH);
    const int nTiles = batch * (MULCH / 16);

    dim3 grid(4096), block(256);  // grid-stride: ~49 tiles/wave amortizes CG fragment load
    hipLaunchKernelGGL(tp_wmma_kernel, grid, block, 0, stream,
                       x1, x2, cg, out, nTiles);
}